// Raycaster_31619549233833
// MI455X (gfx1250) — compile-verified
//
#include <hip/hip_runtime.h>
#include <math.h>

// ---------------- constants mirroring the reference ----------------
#define RW 256          // render width  (u / first output axis)
#define RH 256          // render height (v / second output axis)
#define TF_RES 128
#define MAX_SAMPLES 256
#define NEARP 0.1f
// FOV_RAD = radians(30)
#define FOV_RAD 0.52359877559829887f
// VOL_DIAG = norm((256,256,256)) = 256*sqrt(3)
#define VOL_DIAG 443.40500673763259f

// Builtin expects pointers to int-vector-4 (b128) with address-space qualifiers:
// AS1 (global) source, AS3 (LDS) destination. Go through uintptr_t so the
// 32-bit AS3 pointer carries the LDS byte offset (generic-LDS low bits == offset).
typedef int v4i_t __attribute__((ext_vector_type(4)));
typedef __attribute__((address_space(1))) v4i_t* g_v4i_ptr;
typedef __attribute__((address_space(3))) v4i_t* l_v4i_ptr;
#define AS1_V4I(p) ((g_v4i_ptr)(uintptr_t)(p))
#define AS3_V4I(p) ((l_v4i_ptr)(uintptr_t)(p))

__global__ __launch_bounds__(256)
void raycast_gfx1250(const float* __restrict__ vol,    // 256^3, strides (65536,256,1)
                     const float* __restrict__ tf,     // 128 x 4
                     const float* __restrict__ lf3,    // look_from (3,)
                     float*       __restrict__ out)    // 256 x 256 x 4
{
    // ---- stage transfer function in LDS (2 KB) ----
    __shared__ float4 tf_lds[TF_RES];
    const int tid = threadIdx.y * 16 + threadIdx.x;

#if defined(__gfx1250__) && __has_builtin(__builtin_amdgcn_global_load_async_to_lds_b128)
    // CDNA5 async copy: cache -> LDS directly, no VGPR round trip (ASYNCcnt tracked)
    if (tid < TF_RES) {
        __builtin_amdgcn_global_load_async_to_lds_b128(
            AS1_V4I(tf + 4 * tid),
            AS3_V4I(&tf_lds[tid]),
            /*imm offset*/ 0, /*cpol*/ 0);
    }
#if __has_builtin(__builtin_amdgcn_s_wait_asynccnt)
    __builtin_amdgcn_s_wait_asynccnt(0);
#else
    asm volatile("s_wait_asynccnt 0x0" ::: "memory");
#endif
#else
    if (tid < TF_RES) {
        tf_lds[tid] = ((const float4*)tf)[tid];
    }
#endif
    __syncthreads();

    // thread -> pixel: fastest thread dim = v (H axis, innermost in memory)
    const int py = blockIdx.x * 16 + threadIdx.x;   // v index (H)
    const int px = blockIdx.y * 16 + threadIdx.y;   // u index (W)

    const float lfx = lf3[0], lfy = lf3[1], lfz = lf3[2];

    // ---- camera basis (matches reference exactly; uniform -> scalar pipe) ----
    float inv = 1.0f / sqrtf(lfx*lfx + lfy*lfy + lfz*lfz);
    float vdx = -lfx * inv, vdy = -lfy * inv, vdz = -lfz * inv; // view_dir

    // right = normalize(cross(view_dir, (0,1,0)))
    float rx = vdy*0.0f - vdz*1.0f;
    float ry = vdz*0.0f - vdx*0.0f;
    float rz = vdx*1.0f - vdy*0.0f;
    inv = 1.0f / sqrtf(rx*rx + ry*ry + rz*rz);
    rx *= inv; ry *= inv; rz *= inv;

    // up = normalize(cross(right, view_dir))
    float ux = ry*vdz - rz*vdy;
    float uy = rz*vdx - rx*vdz;
    float uz = rx*vdy - ry*vdx;
    inv = 1.0f / sqrtf(ux*ux + uy*uy + uz*uz);
    ux *= inv; uy *= inv; uz *= inv;

    const float near_h = 2.0f * tanf(FOV_RAD) * NEARP;  // aspect == 1
    const float near_w = near_h;

    const float uu = ((float)px + 0.5f) / (float)RW - 0.5f;
    const float vv = ((float)py + 0.5f) / (float)RH - 0.5f;

    float rdx = NEARP*vdx + uu*near_w*rx + vv*near_h*ux;
    float rdy = NEARP*vdy + uu*near_w*ry + vv*near_h*uy;
    float rdz = NEARP*vdz + uu*near_w*rz + vv*near_h*uz;
    inv = 1.0f / sqrtf(rdx*rdx + rdy*rdy + rdz*rdz);
    rdx *= inv; rdy *= inv; rdz *= inv;

    // ---- slab AABB intersection, box [-1,1]^3 ----
    float sx = (fabsf(rdx) < 1e-9f) ? 1e-9f : rdx;
    float sy = (fabsf(rdy) < 1e-9f) ? 1e-9f : rdy;
    float sz = (fabsf(rdz) < 1e-9f) ? 1e-9f : rdz;
    float dfx = 1.0f / sx, dfy = 1.0f / sy, dfz = 1.0f / sz;
    float t1x = (-1.0f - lfx)*dfx, t2x = (1.0f - lfx)*dfx;
    float t1y = (-1.0f - lfy)*dfy, t2y = (1.0f - lfy)*dfy;
    float t1z = (-1.0f - lfz)*dfz, t2z = (1.0f - lfz)*dfz;
    float tmin = fmaxf(fmaxf(fminf(t1x,t2x), fminf(t1y,t2y)), fminf(t1z,t2z));
    float tmax = fminf(fminf(fmaxf(t1x,t2x), fmaxf(t1y,t2y)), fmaxf(t1z,t2z));
    const bool hit = (tmax >= 0.0f) && (tmin <= tmax);

    float acc_r = 0.0f, acc_g = 0.0f, acc_b = 0.0f, acc_a = 0.0f;

    if (hit) {
        const float entry   = fmaxf(tmin, 0.0f);
        const float ray_len = fmaxf(tmax - entry, 0.0f);
        float nsf = ceilf(VOL_DIAG * ray_len * 0.5f);       // SAMPLING_RATE == 1
        nsf = fminf(fmaxf(nsf, 1.0f), (float)MAX_SAMPLES);
        const int   ns = (int)nsf;
        const float dt = ray_len / nsf;
        const float scale = 255.0f - 1.0e-4f;               // (res-1) - 1e-4

        for (int k = 0; k < ns; ++k) {
            // ---- lookahead prefetch (gfx1250 global_prefetch_b8), 8 samples ahead ----
            {
                float tp  = fmaf((float)k + 8.5f, dt, entry);
                float ppx = fmaf(rdx, tp, lfx);
                float ppy = fmaf(rdy, tp, lfy);
                float ppz = fmaf(rdz, tp, lfz);
                int axl = (int)(fminf(fmaxf(fmaf(ppx, 0.5f, 0.5f), 0.0f), 1.0f) * scale);
                int ayl = (int)(fminf(fmaxf(fmaf(ppy, 0.5f, 0.5f), 0.0f), 1.0f) * scale);
                int azl = (int)(fminf(fmaxf(fmaf(ppz, 0.5f, 0.5f), 0.0f), 1.0f) * scale);
                __builtin_prefetch(vol + ((axl << 16) + (ayl << 8) + azl), 0, 3);
            }

            const float t  = fmaf((float)k + 0.5f, dt, entry);
            const float wx = fmaf(rdx, t, lfx);
            const float wy = fmaf(rdy, t, lfy);
            const float wz = fmaf(rdz, t, lfz);

            // ---- trilinear sample of the volume ----
            const float qx = fminf(fmaxf(fmaf(wx, 0.5f, 0.5f), 0.0f), 1.0f) * scale;
            const float qy = fminf(fmaxf(fmaf(wy, 0.5f, 0.5f), 0.0f), 1.0f) * scale;
            const float qz = fminf(fmaxf(fmaf(wz, 0.5f, 0.5f), 0.0f), 1.0f) * scale;
            const float flx = floorf(qx), fly = floorf(qy), flz = floorf(qz);
            const int xl = (int)flx, yl = (int)fly, zl = (int)flz;
            const int xh = min(xl + 1, 255);
            const int yh = min(yl + 1, 255);
            const int zh = min(zl + 1, 255);
            const float fx = qx - flx, fy = qy - fly, fz = qz - flz;

            const float* b0 = vol + (xl << 16);
            const float* b1 = vol + (xh << 16);
            const int r0 = yl << 8, r1 = yh << 8;

            const float c000 = b0[r0 + zl], c001 = b0[r0 + zh];
            const float c010 = b0[r1 + zl], c011 = b0[r1 + zh];
            const float c100 = b1[r0 + zl], c101 = b1[r0 + zh];
            const float c110 = b1[r1 + zl], c111 = b1[r1 + zh];

            const float c00 = fmaf(fx, c100 - c000, c000);
            const float c01 = fmaf(fx, c101 - c001, c001);
            const float c10 = fmaf(fx, c110 - c010, c010);
            const float c11 = fmaf(fx, c111 - c011, c011);
            const float c0  = fmaf(fy, c10 - c00, c00);
            const float c1  = fmaf(fy, c11 - c01, c01);
            const float intensity = fmaf(fz, c1 - c0, c0);

            // ---- transfer-function lookup (LDS) ----
            const float ti  = fminf(fmaxf(intensity, 0.0f), 1.0f) * (float)(TF_RES - 1);
            const float flt = floorf(ti);
            const int   lo  = (int)flt;
            const int   hi  = min(lo + 1, TF_RES - 1);
            const float fr  = ti - flt;
            const float4 a4 = tf_lds[lo];
            const float4 b4 = tf_lds[hi];
            const float cr = fmaf(fr, b4.x - a4.x, a4.x);
            const float cg = fmaf(fr, b4.y - a4.y, a4.y);
            const float cb = fmaf(fr, b4.z - a4.z, a4.z);
            const float ca = fmaf(fr, b4.w - a4.w, a4.w);

            // opacity correction: SAMPLING_RATE==1 -> pow(x,1) == x exactly
            const float alpha = 1.0f - fmaxf(1.0f - ca, 1e-7f);
            const float w = (1.0f - acc_a) * alpha;
            acc_r = fmaf(w, cr, acc_r);
            acc_g = fmaf(w, cg, acc_g);
            acc_b = fmaf(w, cb, acc_b);
            acc_a += w;

            if (acc_a >= 0.99f) break;   // exact: valid-mask is monotone in reference
        }
    }

    float4 o;
    o.x = acc_r; o.y = acc_g; o.z = acc_b; o.w = acc_a;
    ((float4*)out)[px * RH + py] = o;    // (W,H,4) row-major, coalesced in v
}

extern "C" void kernel_launch(void* const* d_in, const int* in_sizes, int n_in,
                              void* d_out, int out_size, void* d_ws, size_t ws_size,
                              hipStream_t stream) {
    (void)in_sizes; (void)n_in; (void)out_size; (void)d_ws; (void)ws_size;
    const float* vol = (const float*)d_in[0];   // 256^3 fp32
    const float* tf  = (const float*)d_in[1];   // 128x4 fp32
    const float* lf  = (const float*)d_in[2];   // 3 fp32
    float* out = (float*)d_out;                 // 256x256x4 fp32

    dim3 block(16, 16, 1);   // 256 threads = 8 wave32
    dim3 grid(RH / 16, RW / 16, 1);
    raycast_gfx1250<<<grid, block, 0, stream>>>(vol, tf, lf, out);
}